// GCNLayer_48215302865295
// MI455X (gfx1250) — compile-verified
//
#include <hip/hip_runtime.h>
#include <hip/hip_bf16.h>

typedef __attribute__((ext_vector_type(2))) float v2f;
typedef __attribute__((ext_vector_type(8))) float v8f;

#define GCN_K 32
#define GCN_BATCH 64

// One wave32 per node n.
//   out[b,n] = sigmoid( sum_k x_in[b, adj[n,k]] * w[n,k] + bias[n] )
// V_WMMA_F32_16X16X4_F32: M=16 batches, K=4 k-chunk, N=16 replicated columns
// holding the node's weights. Weights/indices/bias ride the scalar (SMEM)
// path; the vector pipe carries only the 64 random x_in gathers (bottleneck)
// plus 4 compacted result stores.
__global__ __launch_bounds__(256)
void gcn_gather_wmma(const float* __restrict__ x_in,
                     const float* __restrict__ weight,
                     const float* __restrict__ bias,
                     const int*   __restrict__ adj,
                     float* __restrict__ out,
                     int n_nodes)
{
    const int wavesPerBlock = blockDim.x >> 5;
    int node = blockIdx.x * wavesPerBlock + (threadIdx.x >> 5);
    if (node >= n_nodes) return;                 // wave-uniform: EXEC stays all-1s
    node = __builtin_amdgcn_readfirstlane(node); // force scalar -> SMEM loads below

    const int lane  = threadIdx.x & 31;
    const int hi    = lane >> 4;                 // 0: K∈{4c,4c+1}, 1: K∈{4c+2,4c+3}
    const int m     = lane & 15;                 // batch-row within the 16x16 tile
    const bool up   = hi != 0;

    const float* wp = weight + (size_t)node * GCN_K;   // uniform -> s_load
    const int*   ap = adj    + (size_t)node * GCN_K;

    // Build B operands (weights broadcast to all 16 columns) and index pairs
    // from scalar data; per-lane-half selection is a one-time v_cndmask each.
    v2f Bc[8];
    int ix0[8], ix1[8];
#pragma unroll
    for (int c = 0; c < 8; ++c) {
        const float w0 = wp[4*c+0], w1 = wp[4*c+1], w2 = wp[4*c+2], w3 = wp[4*c+3];
        const int   a0 = ap[4*c+0], a1 = ap[4*c+1], a2 = ap[4*c+2], a3 = ap[4*c+3];
        Bc[c].x = up ? w2 : w0;
        Bc[c].y = up ? w3 : w1;
        ix0[c]  = up ? a2 : a0;
        ix1[c]  = up ? a3 : a1;
    }
    const float bn = bias[node];

    // Lane-dependent pieces of the epilogue, hoisted out of the g-loop.
    const int  r3 = m & 7;                       // which D row this lane extracts
    const bool s0 = (m & 1) != 0;
    const bool s1 = (m & 2) != 0;
    const bool s2 = (m & 4) != 0;
    const int  rowoff = (r3 + 8 * hi) * n_nodes + node;   // 32-bit store offset

#pragma unroll
    for (int g = 0; g < 4; ++g) {                // 4 groups of 16 batches
        // 32-bit element index (max ~6.5M) -> SADDR + VGPR_I32 addressing
        const int base = (g * 16 + m) * n_nodes;

        v8f acc;
#pragma unroll
        for (int c = 0; c < 8; ++c) {            // K = 32 in chunks of 4
            v2f a;
            a.x = x_in[base + ix0[c]];           // random 4B gathers, L2-resident
            a.y = x_in[base + ix1[c]];
            if (c == 0) {
                // first WMMA: C = inline zero (ISA: SRC2 may be inline 0)
                v8f z = { 0.f, 0.f, 0.f, 0.f, 0.f, 0.f, 0.f, 0.f };
                acc = __builtin_amdgcn_wmma_f32_16x16x4_f32(
                          false, a, false, Bc[0], (short)0, z, false, false);
            } else {
                acc = __builtin_amdgcn_wmma_f32_16x16x4_f32(
                          false, a, false, Bc[c], (short)0, acc, false, false);
            }
        }

        // D row r lives in acc[r]: lanes 0-15 = batch g*16+r, lanes 16-31 =
        // batch g*16+8+r, replicated across columns. Select tree (7 cndmask)
        // gives lane m<8 of each half its own batch value -> ONE 16-lane store.
        const float t0 = s0 ? acc[1] : acc[0];
        const float t1 = s0 ? acc[3] : acc[2];
        const float t2 = s0 ? acc[5] : acc[4];
        const float t3 = s0 ? acc[7] : acc[6];
        const float u0 = s1 ? t1 : t0;
        const float u1 = s1 ? t3 : t2;
        const float vs = (s2 ? u1 : u0) + bn;    // bias added once, fp32

        // fast sigmoid: v_exp_f32 + v_rcp_f32
        const float sg = __builtin_amdgcn_rcpf(1.0f + __expf(-vs));

        if (m < 8) {                             // lanes 0-7 and 16-23 active
            out[rowoff + g * 16 * n_nodes] = sg;
        }
    }
}

extern "C" void kernel_launch(void* const* d_in, const int* in_sizes, int n_in,
                              void* d_out, int out_size, void* d_ws, size_t ws_size,
                              hipStream_t stream)
{
    const float* x_in   = (const float*)d_in[0];   // (64, N) f32
    const float* weight = (const float*)d_in[1];   // (N, 32) f32
    const float* bias   = (const float*)d_in[2];   // (N,)    f32
    const int*   adj    = (const int*)d_in[3];     // (N, 32) int32 (JAX x64-off)

    float* out = (float*)d_out;                    // (64, N) f32

    const int n_nodes = in_sizes[2];               // bias length == N
    const int block = 256;                         // 8 wave32 per block
    const int wavesPerBlock = block / 32;
    const int grid = (n_nodes + wavesPerBlock - 1) / wavesPerBlock;

    gcn_gather_wmma<<<grid, block, 0, stream>>>(x_in, weight, bias, adj, out, n_nodes);
}